// TemplateTransformerEncoder_28991029248254
// MI455X (gfx1250) — compile-verified
//
#include <hip/hip_runtime.h>
#include <hip/hip_bf16.h>
#include <math.h>

// ---------------- problem constants (match reference) ----------------
#define NB     2
#define NN     2048
#define NROWS  (NB * NN)      // 4096
#define NHEADS 8
#define KNN    32
#define BN_EPS 1e-5f

typedef __attribute__((ext_vector_type(2))) float v2f;
typedef __attribute__((ext_vector_type(8))) float v8f;

// =====================================================================
// fp32 WMMA GEMM:  C[m,n] = sum_k A[m,k] * W[n,k] (+ bias[n])
// One wave computes a 16(M) x 64(N) strip = 4 WMMA tiles, K stepped by 4
// (V_WMMA_F32_16X16X4_F32). The A fragment is loaded once per k-quad and
// reused across the 4 N-tiles. The loop is software-pipelined in source:
// fragments for quad k+4 are loaded into fresh registers before the four
// WMMAs for quad k issue, so the in-order VMEM waits become loadcnt<=5
// and load latency overlaps the matrix pipe.
// A-fragment layout (ISA 7.12.2, 32-bit A 16x4):
//   lanes 0-15 : M=lane,    VGPR0=K0, VGPR1=K1
//   lanes16-31 : M=lane-16, VGPR0=K2, VGPR1=K3
// B mirrors with N in place of M. C/D: VGPR r -> row r (lanes<16) / r+8.
// Requires: M % 16 == 0, Nout % 64 == 0, Kdim % 4 == 0 (true for all calls).
// =====================================================================
__global__ __launch_bounds__(128) void gemm_wmma_f32(
    const float* __restrict__ A, int lda,
    const float* __restrict__ W, int ldw,
    const float* __restrict__ bias,
    float* __restrict__ C, int ldc,
    int Mtiles, int Nquads, int Kdim)
{
    const int wave  = blockIdx.x * (blockDim.x >> 5) + (threadIdx.x >> 5);
    const int total = Mtiles * Nquads;
    if (wave >= total) return;                 // wave-uniform; EXEC stays all-ones inside

    const int m0   = (wave / Nquads) * 16;
    const int n0   = (wave % Nquads) * 64;
    const int lane = threadIdx.x & 31;
    const int half = lane >> 4;                // 0: K0/K1, 1: K2/K3
    const int l    = lane & 15;

    // all fragment addresses are 8B-aligned: lda/ldw even, half*2 even
    const float* __restrict__ arow = A + (size_t)(m0 + l) * lda + half * 2;
    const float* __restrict__ wrow = W + (size_t)(n0 + l) * ldw + half * 2;
    const size_t wstep = (size_t)16 * ldw;     // next 16-col group of C = next 16 rows of W

    v8f acc0 = {}, acc1 = {}, acc2 = {}, acc3 = {};

    v2f a  = *(const v2f*)(arow);
    v2f b0 = *(const v2f*)(wrow);
    v2f b1 = *(const v2f*)(wrow + wstep);
    v2f b2 = *(const v2f*)(wrow + 2 * wstep);
    v2f b3 = *(const v2f*)(wrow + 3 * wstep);

    for (int k = 4; k < Kdim; k += 4) {
        // prefetch next k-quad's fragments into fresh registers
        const v2f an  = *(const v2f*)(arow + k);
        const v2f bn0 = *(const v2f*)(wrow + k);
        const v2f bn1 = *(const v2f*)(wrow + k + wstep);
        const v2f bn2 = *(const v2f*)(wrow + k + 2 * wstep);
        const v2f bn3 = *(const v2f*)(wrow + k + 3 * wstep);

        acc0 = __builtin_amdgcn_wmma_f32_16x16x4_f32(false, a, false, b0, (short)0, acc0, false, false);
        acc1 = __builtin_amdgcn_wmma_f32_16x16x4_f32(false, a, false, b1, (short)0, acc1, false, false);
        acc2 = __builtin_amdgcn_wmma_f32_16x16x4_f32(false, a, false, b2, (short)0, acc2, false, false);
        acc3 = __builtin_amdgcn_wmma_f32_16x16x4_f32(false, a, false, b3, (short)0, acc3, false, false);

        a = an; b0 = bn0; b1 = bn1; b2 = bn2; b3 = bn3;
    }
    acc0 = __builtin_amdgcn_wmma_f32_16x16x4_f32(false, a, false, b0, (short)0, acc0, false, false);
    acc1 = __builtin_amdgcn_wmma_f32_16x16x4_f32(false, a, false, b1, (short)0, acc1, false, false);
    acc2 = __builtin_amdgcn_wmma_f32_16x16x4_f32(false, a, false, b2, (short)0, acc2, false, false);
    acc3 = __builtin_amdgcn_wmma_f32_16x16x4_f32(false, a, false, b3, (short)0, acc3, false, false);

    const int mrow = m0 + half * 8;
    float bn0 = 0.f, bn1 = 0.f, bn2 = 0.f, bn3 = 0.f;
    if (bias) {
        bn0 = bias[n0 + l];
        bn1 = bias[n0 + 16 + l];
        bn2 = bias[n0 + 32 + l];
        bn3 = bias[n0 + 48 + l];
    }
#pragma unroll
    for (int r = 0; r < 8; ++r) {
        float* crow = C + (size_t)(mrow + r) * ldc + n0 + l;
        crow[0]  = acc0[r] + bn0;
        crow[16] = acc1[r] + bn1;
        crow[32] = acc2[r] + bn2;
        crow[48] = acc3[r] + bn3;
    }
}

// =====================================================================
// kNN-masked attention. One wave32 per (row, head); lane j owns neighbor
// slot j (K == 32 == wave width). Duplicated neighbor indices are masked
// (only first occurrence kept) to reproduce the boolean column mask.
// QK buffer holds Q at col offset 0 and K at col offset `koff`, ld = ldqk.
// =====================================================================
__global__ __launch_bounds__(256) void attn_kernel(
    const float* __restrict__ QK, int ldqk, int koff,
    const float* __restrict__ V,  int ldv,
    const int*   __restrict__ neigh,
    float* __restrict__ O, int ldo,
    int hd)
{
    const int wid   = blockIdx.x * (blockDim.x >> 5) + (threadIdx.x >> 5);
    const int total = NROWS * NHEADS;
    if (wid >= total) return;

    const int h    = wid % NHEADS;
    const int row  = wid / NHEADS;             // b*N + q
    const int lane = threadIdx.x & 31;
    const int b    = row / NN;

    const int idx  = neigh[(size_t)row * KNN + lane];
    const int krow = b * NN + idx;

    const float* __restrict__ qp = QK + (size_t)row  * ldqk + h * hd;
    const float* __restrict__ kp = QK + (size_t)krow * ldqk + koff + h * hd;

    float s = 0.0f;
    for (int t = 0; t < hd; ++t) s += qp[t] * kp[t];
    s *= rsqrtf((float)hd);

    // dedupe repeated neighbor indices (keep first occurrence)
    bool dup = false;
    for (int j = 0; j < KNN; ++j) {
        const int oi = __shfl(idx, j, 32);
        if (j < lane && oi == idx) dup = true;
    }
    if (dup) s = -__builtin_inff();

    float m = s;
    for (int off = 16; off > 0; off >>= 1) m = fmaxf(m, __shfl_xor(m, off, 32));
    float e = dup ? 0.0f : __expf(s - m);
    float sum = e;
    for (int off = 16; off > 0; off >>= 1) sum += __shfl_xor(sum, off, 32);
    const float attn = e / sum;

    // attn . V : lane t (< hd) owns output channel t of this head
    float acc = 0.0f;
    for (int j = 0; j < KNN; ++j) {
        const float aj = __shfl(attn, j, 32);
        const int   ij = __shfl(idx,  j, 32);
        if (lane < hd) acc += aj * V[(size_t)(b * NN + ij) * ldv + h * hd + lane];
    }
    if (lane < hd) O[(size_t)row * ldo + h * hd + lane] = acc;
}

// =====================================================================
// Sinusoidal position embedding (PositionEmbeddingCoordsSine, scale=1)
// =====================================================================
__global__ __launch_bounds__(256) void pos_sine_kernel(
    const float* __restrict__ xyz, float* __restrict__ pos, int d, int npf)
{
    const int i = blockIdx.x * blockDim.x + threadIdx.x;
    if (i >= NROWS * d) return;
    const int row = i / d, c = i % d;
    float v = 0.0f;
    if (c < 3 * npf) {
        const int axis = c / npf;
        const int j    = c % npf;
        const int p    = j >> 1;
        // 1/dim_t = 10000^(-2p/npf)
        const float invdim = __expf(-9.2103403719761836f * (2.0f * (float)p) / (float)npf);
        const float arg    = xyz[(size_t)row * 3 + axis] * 6.2831853071795865f * invdim;
        v = (j & 1) ? __cosf(arg) : __sinf(arg);
    }
    pos[i] = v;
}

// qin = pos + x   (x read from strided Xbuf, channels [0,d))
__global__ __launch_bounds__(256) void add_x_kernel(
    float* __restrict__ pos, const float* __restrict__ X, int d)
{
    const int i = blockIdx.x * blockDim.x + threadIdx.x;
    if (i >= NROWS * d) return;
    const int row = i / d, c = i % d;
    pos[i] += X[(size_t)row * 256 + c];
}

// sres = x - proj   (residual subtract, per reference)
__global__ __launch_bounds__(256) void sub_kernel(
    const float* __restrict__ X, const float* __restrict__ P,
    float* __restrict__ S, int d)
{
    const int i = blockIdx.x * blockDim.x + threadIdx.x;
    if (i >= NROWS * d) return;
    const int row = i / d, c = i % d;
    S[i] = X[(size_t)row * 256 + c] - P[i];
}

// Z[row,c] = sum_f src[n,b,f] * premlp_w[c,f]   (INFEAT == 3)
__global__ __launch_bounds__(256) void premlp_kernel(
    const float* __restrict__ src, const float* __restrict__ w,
    float* __restrict__ Z)
{
    const int i = blockIdx.x * blockDim.x + threadIdx.x;
    if (i >= NROWS * 64) return;
    const int row = i / 64, c = i % 64;
    const int b = row / NN, n = row % NN;
    const float* __restrict__ sp = src + ((size_t)n * NB + b) * 3;
    Z[i] = sp[0] * w[c * 3 + 0] + sp[1] * w[c * 3 + 1] + sp[2] * w[c * 3 + 2];
}

// Per-channel batch statistics over all NROWS rows (training-mode BN)
__global__ __launch_bounds__(256) void bn_stats_kernel(
    const float* __restrict__ Z, int ldz,
    float* __restrict__ mu, float* __restrict__ rs)
{
    const int c   = blockIdx.x;
    const int tid = threadIdx.x;
    float s1 = 0.0f, s2 = 0.0f;
    for (int r = tid; r < NROWS; r += 256) {
        const float v = Z[(size_t)r * ldz + c];
        s1 += v; s2 += v * v;
    }
    __shared__ float sh1[256], sh2[256];
    sh1[tid] = s1; sh2[tid] = s2;
    __syncthreads();
    for (int off = 128; off > 0; off >>= 1) {
        if (tid < off) { sh1[tid] += sh1[tid + off]; sh2[tid] += sh2[tid + off]; }
        __syncthreads();
    }
    if (tid == 0) {
        const float m   = sh1[0] / (float)NROWS;
        const float var = sh2[0] / (float)NROWS - m * m;   // biased var
        mu[c] = m;
        rs[c] = rsqrtf(var + BN_EPS);
    }
}

// normalize + affine + LeakyReLU(0.2); optional second destination
// (Xbuf concat slot + Cat slot); transposeOut writes (B,C,N) for final out.
__global__ __launch_bounds__(256) void bn_apply_kernel(
    const float* __restrict__ Z, int ldz, int C,
    const float* __restrict__ mu, const float* __restrict__ rs,
    const float* __restrict__ gamma, const float* __restrict__ beta,
    float* __restrict__ dst, int ldd, int doff,
    float* __restrict__ dst2, int ldd2, int d2off,
    int transposeOut)
{
    const int i = blockIdx.x * blockDim.x + threadIdx.x;
    if (i >= NROWS * C) return;
    const int row = i / C, c = i % C;
    float y = (Z[(size_t)row * ldz + c] - mu[c]) * rs[c] * gamma[c] + beta[c];
    y = (y > 0.0f) ? y : 0.2f * y;
    if (transposeOut) {
        const int b = row / NN, n = row % NN;
        dst[((size_t)b * C + c) * NN + n] = y;
    } else {
        dst[(size_t)row * ldd + doff + c] = y;
        if (dst2) dst2[(size_t)row * ldd2 + d2off + c] = y;
    }
}

// =====================================================================
// launch
// =====================================================================
static inline int cdiv(long long a, long long b) { return (int)((a + b - 1) / b); }

extern "C" void kernel_launch(void* const* d_in, const int* in_sizes, int n_in,
                              void* d_out, int out_size, void* d_ws, size_t ws_size,
                              hipStream_t stream)
{
    (void)in_sizes; (void)n_in; (void)out_size; (void)ws_size;

    // inputs in setup_inputs() insertion order
    const float* src      = (const float*)d_in[0];
    const float* src_xyz  = (const float*)d_in[1];
    const int*   neigh    = (const int*)  d_in[2];
    const float* premlp_w = (const float*)d_in[3];
    const float* premlp_g = (const float*)d_in[4];
    const float* premlp_b = (const float*)d_in[5];
    struct Layer { const float *in_w, *in_b, *out_w, *out_b, *lin_w, *gamma, *beta; };
    Layer L[3];
    for (int i = 0; i < 3; ++i) {
        const int k = 6 + i * 7;
        L[i].in_w  = (const float*)d_in[k + 0];
        L[i].in_b  = (const float*)d_in[k + 1];
        L[i].out_w = (const float*)d_in[k + 2];
        L[i].out_b = (const float*)d_in[k + 3];
        L[i].lin_w = (const float*)d_in[k + 4];
        L[i].gamma = (const float*)d_in[k + 5];
        L[i].beta  = (const float*)d_in[k + 6];
    }
    const float* mlp_w = (const float*)d_in[27];
    const float* mlp_g = (const float*)d_in[28];
    const float* mlp_b = (const float*)d_in[29];

    // workspace layout (floats); ~45.1 MB total
    float* ws   = (float*)d_ws;
    float* Xbuf = ws;                              // 4096*256  growing concat x
    float* Cat  = Xbuf + (size_t)NROWS * 256;      // 4096*448  [out0|out1|out2]
    float* Pos  = Cat  + (size_t)NROWS * 448;      // 4096*256  pos / qin / attn-out
    float* QK   = Pos  + (size_t)NROWS * 256;      // 4096*512  [Q|K]
    float* Vb   = QK   + (size_t)NROWS * 512;      // 4096*256  V
    float* Z    = Vb   + (size_t)NROWS * 256;      // 4096*1024 quartered scratch
    float* mu   = Z    + (size_t)NROWS * 1024;     // 1024
    float* rs   = mu   + 1024;                     // 1024
    float* Proj = Z;                               // quarter 0 (4096*256)
    float* Sres = Z + (size_t)NROWS * 256;         // quarter 1
    float* Zl   = Z + (size_t)NROWS * 512;         // quarter 2 (pre-BN layer act)

    const int Mt = NROWS / 16;                     // 256 M-tiles

    // ---- pre-MLP: z = x @ W^T ; BN + LeakyReLU -> Xbuf[:, 0:64) ----
    premlp_kernel<<<cdiv((long long)NROWS * 64, 256), 256, 0, stream>>>(src, premlp_w, Zl);
    bn_stats_kernel<<<64, 256, 0, stream>>>(Zl, 64, mu, rs);
    bn_apply_kernel<<<cdiv((long long)NROWS * 64, 256), 256, 0, stream>>>(
        Zl, 64, 64, mu, rs, premlp_g, premlp_b, Xbuf, 256, 0, nullptr, 0, 0, 0);

    const int dims[3]   = {64, 128, 256};
    const int xoff[3]   = {64, 128, -1};   // where out_i lands in Xbuf
    const int catoff[3] = {0, 64, 192};    // where out_i lands in Cat

    for (int i = 0; i < 3; ++i) {
        const int d   = dims[i];
        const int hd  = d / NHEADS;
        const int npf = (d / 3) / 2 * 2;

        pos_sine_kernel<<<cdiv((long long)NROWS * d, 256), 256, 0, stream>>>(src_xyz, Pos, d, npf);
        add_x_kernel   <<<cdiv((long long)NROWS * d, 256), 256, 0, stream>>>(Pos, Xbuf, d);

        // [Q|K] = qin @ [Wq;Wk]^T + [bq;bk]
        {
            const int Nq = (2 * d) / 64;
            gemm_wmma_f32<<<cdiv((long long)Mt * Nq, 4), 128, 0, stream>>>(
                Pos, d, L[i].in_w, d, L[i].in_b, QK, 2 * d, Mt, Nq, d);
        }
        // V = x @ Wv^T + bv
        {
            const int Nq = d / 64;
            gemm_wmma_f32<<<cdiv((long long)Mt * Nq, 4), 128, 0, stream>>>(
                Xbuf, 256, L[i].in_w + (size_t)2 * d * d, d, L[i].in_b + 2 * d,
                Vb, d, Mt, Nq, d);
        }
        // sparse masked attention -> Attn (reuses Pos)
        attn_kernel<<<cdiv((long long)NROWS * NHEADS * 32, 256), 256, 0, stream>>>(
            QK, 2 * d, d, Vb, d, neigh, Pos, d, hd);
        // Proj = Attn @ out_w^T + out_b
        {
            const int Nq = d / 64;
            gemm_wmma_f32<<<cdiv((long long)Mt * Nq, 4), 128, 0, stream>>>(
                Pos, d, L[i].out_w, d, L[i].out_b, Proj, d, Mt, Nq, d);
        }
        // Sres = x - Proj
        sub_kernel<<<cdiv((long long)NROWS * d, 256), 256, 0, stream>>>(Xbuf, Proj, Sres, d);
        // Zl = Sres @ lin_w^T (no bias)
        {
            const int Nq = d / 64;
            gemm_wmma_f32<<<cdiv((long long)Mt * Nq, 4), 128, 0, stream>>>(
                Sres, d, L[i].lin_w, d, nullptr, Zl, d, Mt, Nq, d);
        }
        // BN + LeakyReLU -> Xbuf concat slot (+ Cat slot)
        bn_stats_kernel<<<d, 256, 0, stream>>>(Zl, d, mu, rs);
        if (i < 2) {
            bn_apply_kernel<<<cdiv((long long)NROWS * d, 256), 256, 0, stream>>>(
                Zl, d, d, mu, rs, L[i].gamma, L[i].beta,
                Xbuf, 256, xoff[i], Cat, 448, catoff[i], 0);
        } else {
            bn_apply_kernel<<<cdiv((long long)NROWS * d, 256), 256, 0, stream>>>(
                Zl, d, d, mu, rs, L[i].gamma, L[i].beta,
                Cat, 448, catoff[i], nullptr, 0, 0, 0);
        }
    }

    // ---- final MLP: Z = Cat @ mlp_w^T ; BN + LeakyReLU ; transpose out ----
    {
        const int Nq = 1024 / 64;
        gemm_wmma_f32<<<cdiv((long long)Mt * Nq, 4), 128, 0, stream>>>(
            Cat, 448, mlp_w, 448, nullptr, Z, 1024, Mt, Nq, 448);
    }
    bn_stats_kernel<<<1024, 256, 0, stream>>>(Z, 1024, mu, rs);
    bn_apply_kernel<<<cdiv((long long)NROWS * 1024, 256), 256, 0, stream>>>(
        Z, 1024, 1024, mu, rs, mlp_g, mlp_b,
        (float*)d_out, 0, 0, nullptr, 0, 0, /*transposeOut=*/1);
}